// SimplestGCN_72533407695322
// MI455X (gfx1250) — compile-verified
//
#include <hip/hip_runtime.h>

#define NN 100000
#define NE 3200000
#define DF 256
#define NC 8
#define NG 64

typedef float v2f __attribute__((ext_vector_type(2)));
typedef float v8f __attribute__((ext_vector_type(8)));

__device__ __forceinline__ void atomic_add_f32(float* p, float v) {
    __hip_atomic_fetch_add(p, v, __ATOMIC_RELAXED, __HIP_MEMORY_SCOPE_AGENT);
}

// ---------------------------------------------------------------------------
// 1) init workspace: deg = 1.0 (self-loop weight), node_out/pool/cnt = 0
// ---------------------------------------------------------------------------
__global__ void init_ws(float* __restrict__ deg, float* __restrict__ node_out,
                        float* __restrict__ pool, float* __restrict__ cnt) {
    int i = blockIdx.x * blockDim.x + threadIdx.x;
    if (i < NN) deg[i] = 1.0f;
    if (i < NN * NC) node_out[i] = 0.0f;
    if (i < NG * NC) pool[i] = 0.0f;
    if (i < NG) cnt[i] = 0.0f;
}

// ---------------------------------------------------------------------------
// 2) degree accumulation at target nodes
// ---------------------------------------------------------------------------
__global__ void accum_deg(const int* __restrict__ col, const float* __restrict__ ew,
                          float* __restrict__ deg) {
    int e = blockIdx.x * blockDim.x + threadIdx.x;
    if (e >= NE) return;
    atomic_add_f32(&deg[col[e]], ew[e]);
}

// ---------------------------------------------------------------------------
// 3) deg -> dinv = rsqrt(deg)   (deg >= 1 always, self-loops)
// ---------------------------------------------------------------------------
__global__ void make_dinv(float* __restrict__ deg) {
    int i = blockIdx.x * blockDim.x + threadIdx.x;
    if (i >= NN) return;
    deg[i] = rsqrtf(deg[i]);
}

// ---------------------------------------------------------------------------
// 4) h = x @ W via V_WMMA_F32_16X16X4_F32
//    One wave computes a 16-node x 16-col tile (cols 8..15 zero padded).
//    256 threads = 8 waves -> 128 nodes per block.
// ---------------------------------------------------------------------------
__global__ void __launch_bounds__(256) gemm_wmma(const float* __restrict__ x,
                                                 const float* __restrict__ W,
                                                 float* __restrict__ h) {
    __shared__ float Wp[DF * 16];   // 256 x 16, cols >= 8 are zero
    int tid = threadIdx.x;
    for (int idx = tid; idx < DF * 16; idx += 256) {
        int r = idx >> 4, n = idx & 15;
        Wp[idx] = (n < NC) ? W[r * NC + n] : 0.0f;
    }
    __syncthreads();

    int wave = tid >> 5;
    int lane = tid & 31;
    int base = blockIdx.x * 128 + wave * 16;

    int m = lane & 15;                 // A: row within tile / B,D: column
    int khalf = (lane >> 4) << 1;      // lanes 0-15: K+0,K+1 ; lanes 16-31: K+2,K+3

    long node = base + m;
    if (node > (long)(NN - 1)) node = NN - 1;   // clamp tail (stores guarded)
    const float* xrow = x + node * DF;

    v8f acc = {};
    for (int k = 0; k < DF; k += 4) {
        int kk = k + khalf;
        __builtin_prefetch(xrow + kk + 32, 0, 0);          // global_prefetch_b8
        v2f a = *(const v2f*)(xrow + kk);                  // A[m][kk], A[m][kk+1]
        v2f bf;
        bf.x = Wp[kk * 16 + m];                            // B[kk][n]
        bf.y = Wp[(kk + 1) * 16 + m];                      // B[kk+1][n]
        acc = __builtin_amdgcn_wmma_f32_16x16x4_f32(
            /*neg_a=*/false, a, /*neg_b=*/false, bf,
            /*c_mod=*/(short)0, acc, /*reuse_a=*/false, /*reuse_b=*/false);
    }

    // D layout: VGPR r -> lanes 0-15: M=r, N=lane ; lanes 16-31: M=8+r, N=lane-16
    int mofs = (lane < 16) ? 0 : 8;
#pragma unroll
    for (int r = 0; r < 8; ++r) {
        int nodeo = base + mofs + r;
        if (nodeo < NN && m < NC) h[(long)nodeo * NC + m] = acc[r];
    }
}

// ---------------------------------------------------------------------------
// 5) edge scatter: node_out[col] += h[row] * (dinv[row]*w*dinv[col])
// ---------------------------------------------------------------------------
__global__ void edge_scatter(const int* __restrict__ row, const int* __restrict__ col,
                             const float* __restrict__ ew, const float* __restrict__ dinv,
                             const float* __restrict__ h, float* __restrict__ node_out) {
    int e = blockIdx.x * blockDim.x + threadIdx.x;
    if (e >= NE) return;
    int r = row[e], c = col[e];
    float norm = dinv[r] * ew[e] * dinv[c];
    const float4* hr = (const float4*)(h + (long)r * NC);
    float4 h0 = hr[0], h1 = hr[1];
    float* dst = node_out + (long)c * NC;
    atomic_add_f32(dst + 0, h0.x * norm);
    atomic_add_f32(dst + 1, h0.y * norm);
    atomic_add_f32(dst + 2, h0.z * norm);
    atomic_add_f32(dst + 3, h0.w * norm);
    atomic_add_f32(dst + 4, h1.x * norm);
    atomic_add_f32(dst + 5, h1.y * norm);
    atomic_add_f32(dst + 6, h1.z * norm);
    atomic_add_f32(dst + 7, h1.w * norm);
}

// ---------------------------------------------------------------------------
// 6) self-loop contribution + per-graph pooling sums
// ---------------------------------------------------------------------------
__global__ void pool_accum(const float* __restrict__ h, const float* __restrict__ node_out,
                           const float* __restrict__ dinv, const int* __restrict__ batch,
                           float* __restrict__ pool, float* __restrict__ cnt) {
    int i = blockIdx.x * blockDim.x + threadIdx.x;
    if (i >= NN) return;
    float di = dinv[i];
    float sl = di * di;                 // self-loop norm: dinv*1*dinv
    int g = batch[i];
    const float4* hr = (const float4*)(h + (long)i * NC);
    const float4* orw = (const float4*)(node_out + (long)i * NC);
    float4 h0 = hr[0], h1 = hr[1];
    float4 o0 = orw[0], o1 = orw[1];
    float* pg = pool + (long)g * NC;
    atomic_add_f32(pg + 0, o0.x + h0.x * sl);
    atomic_add_f32(pg + 1, o0.y + h0.y * sl);
    atomic_add_f32(pg + 2, o0.z + h0.z * sl);
    atomic_add_f32(pg + 3, o0.w + h0.w * sl);
    atomic_add_f32(pg + 4, o1.x + h1.x * sl);
    atomic_add_f32(pg + 5, o1.y + h1.y * sl);
    atomic_add_f32(pg + 6, o1.z + h1.z * sl);
    atomic_add_f32(pg + 7, o1.w + h1.w * sl);
    atomic_add_f32(&cnt[g], 1.0f);
}

// ---------------------------------------------------------------------------
// 7) finalize: mean pool + bias
// ---------------------------------------------------------------------------
__global__ void finalize(const float* __restrict__ pool, const float* __restrict__ cnt,
                         const float* __restrict__ b, float* __restrict__ out) {
    int i = blockIdx.x * blockDim.x + threadIdx.x;
    if (i >= NG * NC) return;
    int g = i >> 3, j = i & 7;
    float c = cnt[g];
    if (c < 1.0f) c = 1.0f;
    out[i] = pool[i] / c + b[j];
}

extern "C" void kernel_launch(void* const* d_in, const int* in_sizes, int n_in,
                              void* d_out, int out_size, void* d_ws, size_t ws_size,
                              hipStream_t stream) {
    const float* x     = (const float*)d_in[0];   // [N, 256]
    const int*   ei    = (const int*)d_in[1];     // [2, E]
    const float* ew    = (const float*)d_in[2];   // [E]
    const int*   batch = (const int*)d_in[3];     // [N]
    const float* W     = (const float*)d_in[4];   // [256, 8]
    const float* b     = (const float*)d_in[5];   // [8]
    float* out = (float*)d_out;                   // [64, 8]

    float* ws       = (float*)d_ws;
    float* deg      = ws;                         // N floats (becomes dinv in place)
    float* h        = deg + NN;                   // N*8
    float* node_out = h + (long)NN * NC;          // N*8
    float* pool     = node_out + (long)NN * NC;   // 64*8
    float* cnt      = pool + NG * NC;             // 64

    const int* row = ei;
    const int* col = ei + NE;

    init_ws<<<(NN * NC + 255) / 256, 256, 0, stream>>>(deg, node_out, pool, cnt);
    accum_deg<<<(NE + 255) / 256, 256, 0, stream>>>(col, ew, deg);
    make_dinv<<<(NN + 255) / 256, 256, 0, stream>>>(deg);
    gemm_wmma<<<(NN + 127) / 128, 256, 0, stream>>>(x, W, h);
    edge_scatter<<<(NE + 255) / 256, 256, 0, stream>>>(row, col, ew, deg, h, node_out);
    pool_accum<<<(NN + 255) / 256, 256, 0, stream>>>(h, node_out, deg, batch, pool, cnt);
    finalize<<<(NG * NC + 255) / 256, 256, 0, stream>>>(pool, cnt, b, out);
}